// LNNet_36773509989088
// MI455X (gfx1250) — compile-verified
//
#include <hip/hip_runtime.h>

// ---------------------------------------------------------------------------
// Problem constants (match reference):  B=2, N=2048, M=64, F=128, H1=H2=64
// ---------------------------------------------------------------------------
#define BB 2
#define NN 2048
#define MM 64
#define FF 128
#define HH1 64
#define HH2 64
#define FORCE_SCALE 1e10f

typedef __attribute__((ext_vector_type(2))) float v2f;
typedef __attribute__((ext_vector_type(8))) float v8f;

__device__ __forceinline__ float sigmoidf(float x) {
    return 1.0f / (1.0f + __expf(-x));
}

// D = A(16x4 f32) * B(4x16 f32) + C(16x16 f32)  -- CDNA5 wave32 WMMA
__device__ __forceinline__ v8f wmma_f32(v2f a, v2f b, v8f c) {
    return __builtin_amdgcn_wmma_f32_16x16x4_f32(
        /*neg_a=*/false, a, /*neg_b=*/false, b,
        /*c_mod=*/(short)0, c, /*reuse_a=*/false, /*reuse_b=*/false);
}

// ---------------------------------------------------------------------------
// Kernel 0: zero the per-batch energy accumulators (d_out is poisoned 0xAA).
// ---------------------------------------------------------------------------
__global__ void init_etot_kernel(float* __restrict__ etot) {
    if (threadIdx.x < BB) etot[threadIdx.x] = 0.0f;
}

// ---------------------------------------------------------------------------
// Kernel 1: per-atom sigmoid MLP forward + analytic Jacobian via WMMA.
// One wave32 per 16-atom tile.  All GEMMs use V_WMMA_F32_16X16X4_F32.
//   fragment addressing (ISA 7.12.2):
//     A 16x4  : lane.lo = M row,  a = {A[lo][k+2*hi], A[lo][k+2*hi+1]}
//     B 4x16  : lane.lo = N col,  b = {B[k+2*hi][n],  B[k+2*hi+1][n]}
//     C 16x16 : element (M = r + 8*hi, N = 16*t + lo) in zt[t][r]
// ---------------------------------------------------------------------------
__global__ void __launch_bounds__(32)
mlp_grad_kernel(const float* __restrict__ image,   // [B*N, F]
                const float* __restrict__ W0,      // [H1, F]
                const float* __restrict__ W1,      // [H2, H1]
                const float* __restrict__ Wout,    // [1, H2]
                float* __restrict__ g_out,         // [B*N, F] workspace
                float* __restrict__ etot)          // [B]
{
    __shared__ float sh0[16 * HH1];   // h0 activations (row-major 16 x 64)
    __shared__ float sh1[16 * HH2];   // h1 activations
    __shared__ float sht[16 * HH1];   // staging for backprop A-matrices

    const int lane = threadIdx.x;
    const int lo   = lane & 15;
    const int hi   = lane >> 4;
    const int ko   = 2 * hi;
    const int atom0 = blockIdx.x * 16;            // 16 atoms per tile
    const float* A = image + (size_t)atom0 * FF;

    const v8f vz = {0.f, 0.f, 0.f, 0.f, 0.f, 0.f, 0.f, 0.f};

    // ---- layer 0: Z0 = image_tile @ W0^T   (16x128)@(128x64)
    v8f z0[4];
#pragma unroll
    for (int t = 0; t < 4; ++t) z0[t] = vz;
    for (int k = 0; k < FF; k += 4) {
        v2f a;
        a.x = A[lo * FF + k + ko];
        a.y = A[lo * FF + k + ko + 1];
#pragma unroll
        for (int t = 0; t < 4; ++t) {
            const int n = t * 16 + lo;
            v2f b;                                  // B[kk][n] = W0[n][kk]
            b.x = W0[n * FF + k + ko];
            b.y = W0[n * FF + k + ko + 1];
            z0[t] = wmma_f32(a, b, z0[t]);
        }
    }
#pragma unroll
    for (int t = 0; t < 4; ++t)
#pragma unroll
        for (int r = 0; r < 8; ++r)
            sh0[(r + 8 * hi) * HH1 + t * 16 + lo] = sigmoidf(z0[t][r]);
    __syncthreads();   // single wave: S_NOP; DScnt orders LDS within the wave

    // ---- layer 1: Z1 = h0 @ W1^T   (16x64)@(64x64)
    v8f z1[4];
#pragma unroll
    for (int t = 0; t < 4; ++t) z1[t] = vz;
    for (int k = 0; k < HH1; k += 4) {
        v2f a;
        a.x = sh0[lo * HH1 + k + ko];
        a.y = sh0[lo * HH1 + k + ko + 1];
#pragma unroll
        for (int t = 0; t < 4; ++t) {
            const int n = t * 16 + lo;
            v2f b;                                  // B[kk][n] = W1[n][kk]
            b.x = W1[n * HH1 + k + ko];
            b.y = W1[n * HH1 + k + ko + 1];
            z1[t] = wmma_f32(a, b, z1[t]);
        }
    }
#pragma unroll
    for (int t = 0; t < 4; ++t)
#pragma unroll
        for (int r = 0; r < 8; ++r)
            sh1[(r + 8 * hi) * HH2 + t * 16 + lo] = sigmoidf(z1[t][r]);
    __syncthreads();

    // ---- Ei = h1 @ Wout^T, accumulate Etot per batch (tile is batch-uniform)
    if (lane < 16) {
        float e = 0.f;
        for (int n = 0; n < HH2; ++n) e += sh1[lane * HH2 + n] * Wout[n];
        atomicAdd(&etot[atom0 / NN], e);
    }

    // ---- backprop stage 1: t1 = h1*(1-h1)*Wout  (elementwise, A-layout in LDS)
    for (int i = lane; i < 16 * HH2; i += 32) {
        const float h = sh1[i];
        sht[i] = h * (1.f - h) * Wout[i & (HH2 - 1)];
    }
    __syncthreads();

    // ---- g1 = t1 @ W1   (16x64)@(64x64), B[kk][n] = W1[kk][n]
    v8f zg[4];
#pragma unroll
    for (int t = 0; t < 4; ++t) zg[t] = vz;
    for (int k = 0; k < HH2; k += 4) {
        v2f a;
        a.x = sht[lo * HH2 + k + ko];
        a.y = sht[lo * HH2 + k + ko + 1];
#pragma unroll
        for (int t = 0; t < 4; ++t) {
            const int n = t * 16 + lo;
            v2f b;
            b.x = W1[(k + ko) * HH1 + n];
            b.y = W1[(k + ko + 1) * HH1 + n];
            zg[t] = wmma_f32(a, b, zg[t]);
        }
    }
    __syncthreads();

    // ---- backprop stage 2: A2 = h0*(1-h0)*g1  (C-layout -> LDS row-major)
#pragma unroll
    for (int t = 0; t < 4; ++t)
#pragma unroll
        for (int r = 0; r < 8; ++r) {
            const int m = r + 8 * hi;
            const int n = t * 16 + lo;
            const float h = sh0[m * HH1 + n];
            sht[m * HH1 + n] = h * (1.f - h) * zg[t][r];
        }
    __syncthreads();

    // ---- g0 = A2 @ W0   (16x64)@(64x128), B[kk][n] = W0[kk][n], 8 N-tiles
    v8f zo[8];
#pragma unroll
    for (int t = 0; t < 8; ++t) zo[t] = vz;
    for (int k = 0; k < HH1; k += 4) {
        v2f a;
        a.x = sht[lo * HH1 + k + ko];
        a.y = sht[lo * HH1 + k + ko + 1];
#pragma unroll
        for (int t = 0; t < 8; ++t) {
            const int n = t * 16 + lo;
            v2f b;
            b.x = W0[(k + ko) * FF + n];
            b.y = W0[(k + ko + 1) * FF + n];
            zo[t] = wmma_f32(a, b, zo[t]);
        }
    }
#pragma unroll
    for (int t = 0; t < 8; ++t)
#pragma unroll
        for (int r = 0; r < 8; ++r)
            g_out[(size_t)(atom0 + r + 8 * hi) * FF + t * 16 + lo] = zo[t][r];
}

// ---------------------------------------------------------------------------
// Kernel 2: force contraction -- the memory-bound stage (403 MB of dfeat,
// 0.5 FLOP/byte -> HBM roofline ~17us at 23.3 TB/s).  One 256-thread block
// per (b,n).  Coalesced 12B/lane non-temporal streaming of dfeat (keep the
// 2 MB gradient table hot in L2), gather g via LDS-cached neighbor IDs,
// LDS tree reduction to 3 components.
// ---------------------------------------------------------------------------
__global__ void __launch_bounds__(256)
force_kernel(const float* __restrict__ g,         // [B*N, F]
             const float* __restrict__ dfeat,     // [B,N,M,F,3]
             const int*   __restrict__ neighbor,  // [B,N,M], 1-based, 0=pad
             float* __restrict__ force)           // [B*N, 3]
{
    __shared__ int   s_idx[MM];
    __shared__ float red[3][256];

    const int bn  = blockIdx.x;          // 0 .. B*N-1
    const int b   = bn >> 11;            // /NN
    const int tid = threadIdx.x;

    if (tid < MM) s_idx[tid] = neighbor[(size_t)bn * MM + tid];
    __syncthreads();

    const float* df = dfeat + (size_t)bn * MM * FF * 3;
    const float* gb = g + (size_t)b * NN * FF;

    float f0 = 0.f, f1 = 0.f, f2 = 0.f;
    for (int i = tid; i < MM * FF; i += 256) {
        const int m  = i >> 7;           // /FF
        const int ff = i & (FF - 1);
        const int nb = s_idx[m];
        const float gv = (nb > 0) ? gb[(size_t)(nb - 1) * FF + ff] : 0.0f;
        const float* p = df + (size_t)i * 3;
        if (i + 2 * 256 < MM * FF)       // stay ahead of the HBM stream
            __builtin_prefetch(p + (size_t)2 * 256 * 3, 0, 0);
        f0 += gv * __builtin_nontemporal_load(p);
        f1 += gv * __builtin_nontemporal_load(p + 1);
        f2 += gv * __builtin_nontemporal_load(p + 2);
    }

    red[0][tid] = f0; red[1][tid] = f1; red[2][tid] = f2;
    __syncthreads();
#pragma unroll
    for (int s = 128; s > 0; s >>= 1) {
        if (tid < s) {
            red[0][tid] += red[0][tid + s];
            red[1][tid] += red[1][tid + s];
            red[2][tid] += red[2][tid + s];
        }
        __syncthreads();
    }
    if (tid == 0) {
        force[(size_t)bn * 3 + 0] = red[0][0] * FORCE_SCALE;
        force[(size_t)bn * 3 + 1] = red[1][0] * FORCE_SCALE;
        force[(size_t)bn * 3 + 2] = red[2][0] * FORCE_SCALE;
    }
}

// ---------------------------------------------------------------------------
// Host-side launcher.  Inputs (setup_inputs order):
//   0 image [B,N,F]  1 dfeat [B,N,M,F,3]  2 neighbor [B,N,M]
//   3 Egroup_weight (unused)  4 divider (unused)
//   5 W0 [H1,F]  6 W1 [H2,H1]  7 Wout [1,H2]
// Output: Etot [B,1] ++ force [B,N,3]  (floats, concatenated)
// Workspace: gradient table g [B*N,F] = 2 MB at d_ws.
// ---------------------------------------------------------------------------
extern "C" void kernel_launch(void* const* d_in, const int* in_sizes, int n_in,
                              void* d_out, int out_size, void* d_ws, size_t ws_size,
                              hipStream_t stream) {
    (void)in_sizes; (void)n_in; (void)out_size; (void)ws_size;
    const float* image    = (const float*)d_in[0];
    const float* dfeat    = (const float*)d_in[1];
    const int*   neighbor = (const int*)d_in[2];
    const float* W0       = (const float*)d_in[5];
    const float* W1       = (const float*)d_in[6];
    const float* Wout     = (const float*)d_in[7];

    float* out   = (float*)d_out;
    float* etot  = out;          // [B]
    float* force = out + BB;     // [B*N*3]
    float* g     = (float*)d_ws; // [B*N*F] floats (2 MB)

    hipLaunchKernelGGL(init_etot_kernel, dim3(1), dim3(32), 0, stream, etot);
    hipLaunchKernelGGL(mlp_grad_kernel, dim3((BB * NN) / 16), dim3(32), 0, stream,
                       image, W0, W1, Wout, g, etot);
    hipLaunchKernelGGL(force_kernel, dim3(BB * NN), dim3(256), 0, stream,
                       g, dfeat, neighbor, force);
}